// CrossMerge_15891378995283
// MI455X (gfx1250) — compile-verified
//
#include <hip/hip_runtime.h>

// ---------------------------------------------------------------------------
// CrossMerge fused kernel for MI455X (gfx1250)
//   means = mean_C(g_i) -> 1x1 conv(4->4) -> softmax over 4 -> weighted sum
// Bandwidth-bound: 302 MB in + 75.5 MB out => ~16.2 us floor @ 23.3 TB/s.
// Strategy: stage each 128 KB tile (32 pixels x 256 ch x 4 grids) in LDS via
// gfx1250 async global->LDS loads (double buffered), so each grid element is
// read from HBM exactly once; second (weighted-sum) pass reads LDS.
// ---------------------------------------------------------------------------

#define B_      8
#define C_      256
#define HW_     9216            // 96*96, contiguous per (b,c) plane
#define PIX     32              // pixels per tile (consecutive flat hw)
#define TPB     288             // tiles per batch image = HW_/PIX
#define NTILES  2304            // B_*TPB
#define NBLK    576             // NTILES % NBLK == 0 -> 4 tiles per block
#define NTHR    256             // 8 wave32s
#define BUF     (4 * C_ * PIX)  // 32768 floats = 128 KB per tile buffer
#define SCR_OFF (2 * BUF)
#define SMEM_FLOATS (2 * BUF + 4 * 8 * 32)   // 2 buffers + partial-sum scratch

#if defined(__HIP_DEVICE_COMPILE__) && \
    __has_builtin(__builtin_amdgcn_global_load_async_to_lds_b128) && \
    __has_builtin(__builtin_amdgcn_s_wait_asynccnt)
#define USE_ASYNC 1
#else
#define USE_ASYNC 0
#if defined(__HIP_DEVICE_COMPILE__)
#warning "gfx1250 async-to-LDS builtins unavailable; using synchronous fallback"
#endif
#endif

// The builtin's parameter type (from the clang diagnostic) is a pointer to a
// 16-byte int vector: int __attribute__((vector_size(16))).
typedef int v4i_b128 __attribute__((vector_size(4 * sizeof(int))));
typedef __attribute__((address_space(1))) v4i_b128* gptr_b128;
typedef __attribute__((address_space(3))) v4i_b128* lptr_b128;

__device__ __forceinline__ void copy16_g2l(const float* g, float* l) {
#if USE_ASYNC
  // GLOBAL_LOAD_ASYNC_TO_LDS_B128: 16 bytes/lane, tracked by ASYNCcnt.
  __builtin_amdgcn_global_load_async_to_lds_b128(
      (gptr_b128)(g), (lptr_b128)(l), /*offset=*/0, /*cpol=*/0);
#else
  float4 v = *(const float4*)g;
  *(float4*)l = v;
#endif
}

template <int N>
__device__ __forceinline__ void wait_async_le() {
#if USE_ASYNC
  __builtin_amdgcn_s_wait_asynccnt(N);   // wait ASYNCcnt <= N (in-order loads)
#endif
  asm volatile("" ::: "memory");
}

// Wave w async-loads ONLY its own channel slab [32w, 32w+32) of all 4 grids:
// lane l -> channel offset l>>3, pixel offset (l&7)*4; 8 issues per grid.
// Per wave per tile: 32 async b128 issues (fits the 6-bit ASYNCcnt window).
__device__ __forceinline__ void issue_tile(
    const float* g0, const float* g1, const float* g2, const float* g3,
    float* buf, int tile, int wave, int lane) {
  const int b  = tile / TPB;
  const int q0 = (tile - b * TPB) * PIX;
  const int c0 = (wave << 5) + (lane >> 3);
  const int p  = (lane & 7) << 2;
  const size_t gofs = (size_t)(b * C_ + c0) * HW_ + (size_t)(q0 + p);
  const int    lofs = c0 * PIX + p;
  const float* gs[4] = { g0, g1, g2, g3 };
#pragma unroll
  for (int i = 0; i < 4; ++i) {
    const float* gp = gs[i] + gofs;
    float*       lp = buf + i * (C_ * PIX) + lofs;
#pragma unroll
    for (int j = 0; j < 8; ++j) {
      copy16_g2l(gp + (size_t)(j * 4) * HW_, lp + (j * 4) * PIX);
    }
  }
}

__global__ __launch_bounds__(NTHR, 1) void crossmerge_fused_kernel(
    const float* __restrict__ g0, const float* __restrict__ g1,
    const float* __restrict__ g2, const float* __restrict__ g3,
    const float* __restrict__ wproj, const float* __restrict__ bproj,
    float* __restrict__ out) {
  __shared__ float smem[SMEM_FLOATS];
  float* scr = smem + SCR_OFF;

  const int tid  = (int)threadIdx.x;
  const int wave = tid >> 5;
  const int lane = tid & 31;

  // Tiny uniform 4x4 projection weights + bias (L2-hot scalarizable loads).
  float Wm[4][4], Bv[4];
#pragma unroll
  for (int o = 0; o < 4; ++o) {
    Bv[o] = bproj[o];
#pragma unroll
    for (int i = 0; i < 4; ++i) Wm[o][i] = wproj[o * 4 + i];
  }

  int t = (int)blockIdx.x;           // NBLK divides NTILES: 4 tiles per block
  issue_tile(g0, g1, g2, g3, smem, t, wave, lane);

  int parity = 0;
  for (; t < NTILES; t += NBLK) {
    float* cur = smem + parity * BUF;
    const int tn = t + NBLK;
    if (tn < NTILES) {
      issue_tile(g0, g1, g2, g3, smem + (parity ^ 1) * BUF, tn, wave, lane);
      wait_async_le<32>();           // 32 newer issues outstanding => tile t done
    } else {
      wait_async_le<0>();
    }

    // Pass 1: per-wave partial channel sums over OWN slab (no barrier needed
    // for tile data -- each wave consumes only what it loaded itself).
    const int cbase = wave << 5;
    float s0 = 0.f, s1 = 0.f, s2 = 0.f, s3 = 0.f;
    const float* l0 = cur + (0 * C_ + cbase) * PIX + lane;
    const float* l1 = cur + (1 * C_ + cbase) * PIX + lane;
    const float* l2 = cur + (2 * C_ + cbase) * PIX + lane;
    const float* l3 = cur + (3 * C_ + cbase) * PIX + lane;
#pragma unroll
    for (int j = 0; j < 32; ++j) {
      s0 += l0[j * PIX];
      s1 += l1[j * PIX];
      s2 += l2[j * PIX];
      s3 += l3[j * PIX];
    }
    scr[(0 * 8 + wave) * 32 + lane] = s0;
    scr[(1 * 8 + wave) * 32 + lane] = s1;
    scr[(2 * 8 + wave) * 32 + lane] = s2;
    scr[(3 * 8 + wave) * 32 + lane] = s3;
    __syncthreads();

    // Cross-wave reduction (redundant per wave, trivial cost), projection,
    // stable softmax for this lane's pixel.
    float mean[4];
#pragma unroll
    for (int i = 0; i < 4; ++i) {
      float a = 0.f;
#pragma unroll
      for (int w2 = 0; w2 < 8; ++w2) a += scr[(i * 8 + w2) * 32 + lane];
      mean[i] = a * (1.0f / 256.0f);
    }
    float lg[4];
#pragma unroll
    for (int o = 0; o < 4; ++o) {
      lg[o] = Bv[o] + Wm[o][0] * mean[0] + Wm[o][1] * mean[1]
                    + Wm[o][2] * mean[2] + Wm[o][3] * mean[3];
    }
    const float mx = fmaxf(fmaxf(lg[0], lg[1]), fmaxf(lg[2], lg[3]));
    const float e0 = expf(lg[0] - mx), e1 = expf(lg[1] - mx);
    const float e2 = expf(lg[2] - mx), e3 = expf(lg[3] - mx);
    const float inv = 1.0f / (e0 + e1 + e2 + e3);
    const float w0 = e0 * inv, w1 = e1 * inv, w2 = e2 * inv, w3 = e3 * inv;

    // Pass 2: weighted sum from LDS (own slab), nontemporal coalesced stores.
    const int b  = t / TPB;
    const int q0 = (t - b * TPB) * PIX;
    const size_t obase = (size_t)(b * C_ + cbase) * HW_ + (size_t)(q0 + lane);
#pragma unroll
    for (int j = 0; j < 32; ++j) {
      const int ci = cbase + j;
      const float v = w0 * cur[(0 * C_ + ci) * PIX + lane]
                    + w1 * cur[(1 * C_ + ci) * PIX + lane]
                    + w2 * cur[(2 * C_ + ci) * PIX + lane]
                    + w3 * cur[(3 * C_ + ci) * PIX + lane];
      __builtin_nontemporal_store(v, out + obase + (size_t)j * HW_);
    }
    __syncthreads();   // protect scratch before next tile's partial writes
    parity ^= 1;
  }
}

extern "C" void kernel_launch(void* const* d_in, const int* in_sizes, int n_in,
                              void* d_out, int out_size, void* d_ws, size_t ws_size,
                              hipStream_t stream) {
  (void)in_sizes; (void)n_in; (void)out_size; (void)d_ws; (void)ws_size;
  const float* g0 = (const float*)d_in[0];
  const float* g1 = (const float*)d_in[1];
  const float* g2 = (const float*)d_in[2];
  const float* g3 = (const float*)d_in[3];
  const float* wp = (const float*)d_in[4];
  const float* bp = (const float*)d_in[5];
  float* out = (float*)d_out;
  crossmerge_fused_kernel<<<dim3(NBLK), dim3(NTHR), 0, stream>>>(
      g0, g1, g2, g3, wp, bp, out);
}